// GraphVAE_78718160601810
// MI455X (gfx1250) — compile-verified
//
#include <hip/hip_runtime.h>

#define N       96
#define LDW     100        // padded LDS row stride (floats); 400B rows stay 16B-aligned
#define NT      1024       // 32 wave32s -> 8 waves per SIMD32 on one WGP
#define NWAVES  32
#define ITERS   50

typedef __attribute__((ext_vector_type(2))) float v2f;
typedef __attribute__((ext_vector_type(8))) float v8f;

__global__ __launch_bounds__(NT)
void graphvae_mpm_kernel(const float* __restrict__ adj,
                         const float* __restrict__ adj_recon,
                         float* __restrict__ out)
{
    // ~151.6 KB static LDS: fits the CDNA5 WGP's 320 KB easily.
    __shared__ __align__(16) float A_s[N][LDW];   // A' (zero diagonal)
    __shared__ __align__(16) float B_s[N][LDW];   // B' (zero diagonal)
    __shared__ __align__(16) float X_s[N][LDW];   // current x
    __shared__ __align__(16) float M_s[N][LDW];   // M[j][a] = max_b x[j,b]*B'[a,b]
    __shared__ float f_v[N], fr_v[N], d_v[N], dr_v[N];
    __shared__ float partial[NWAVES];

    const int tid  = threadIdx.x;
    const int wave = tid >> 5;
    const int lane = tid & 31;
    const int row  = lane & 15;   // WMMA: M (A-frag) or N (B/C/D-frag) index
    const int hi   = lane >> 4;   // WMMA: K-half (A/B) or +8 row offset (C/D)

    // ---------------- setup: raw loads ----------------
    for (int e = tid; e < N * N; e += NT) {
        const int i = e / N, j = e % N;
        A_s[i][j] = adj[e];
        B_s[i][j] = adj_recon[e];
    }
    __syncthreads();

    // degrees + diagonals
    if (tid < N) {
        float s = 0.f, sr = 0.f;
        for (int j = 0; j < N; ++j) { s += A_s[tid][j]; sr += B_s[tid][j]; }
        f_v[tid]  = s;
        fr_v[tid] = sr;
        d_v[tid]  = A_s[tid][tid];
        dr_v[tid] = B_s[tid][tid];
    }
    __syncthreads();

    // in-place transform to A', B'; init x = 1/N
    for (int e = tid; e < N * N; e += NT) {
        const int i = e / N, j = e % N;
        const float a = (i == j) ? 0.f : A_s[i][j] * d_v[i] * d_v[j];
        const float b = (i == j) ? 0.f : B_s[i][j] * dr_v[i] * dr_v[j];
        A_s[i][j] = a;
        B_s[i][j] = b;
        X_s[i][j] = 1.0f / (float)N;
    }

    const int  t0  = wave;                  // first tile owned by this wave
    const bool two = (wave < 36 - NWAVES);  // waves 0..3 also own tile wave+32

    // ---------------- 50 mpm iterations, fully LDS-resident ----------------
    for (int it = 0; it < ITERS; ++it) {
        __syncthreads();   // B1: x/partial consumers of prev iter done

        // ---- phase 1: M[j][a] = max_b X[j][b] * B'[a][b]  (VALU max-contract) ----
        for (int e = tid; e < N * N; e += NT) {
            const int j = e / N, a = e % N;
            const float4* xr = reinterpret_cast<const float4*>(&X_s[j][0]);
            const float4* br = reinterpret_cast<const float4*>(&B_s[a][0]);
            float m = 0.f;                      // all terms non-negative
            #pragma unroll 4
            for (int q = 0; q < N / 4; ++q) {
                const float4 xv = xr[q];
                const float4 bv = br[q];
                m = fmaxf(m, xv.x * bv.x);
                m = fmaxf(m, xv.y * bv.y);
                m = fmaxf(m, xv.z * bv.z);
                m = fmaxf(m, xv.w * bv.w);
            }
            M_s[j][a] = m;
        }
        __syncthreads();   // B2: M ready

        // ---- phase 2: neigh = A' @ M via V_WMMA_F32_16X16X4_F32; y = x*D + neigh ----
        float ss = 0.f;

        auto doTile = [&](int t) -> v8f {
            const int i0 = (t / 6) * 16;
            const int a0 = (t % 6) * 16;
            v8f c = {};
            #pragma unroll
            for (int kb = 0; kb < N; kb += 4) {
                v2f av, bv;
                // A 16x4 frag: lane(row, K-half); adjacent pair -> ds_load_b64
                av.x = A_s[i0 + row][kb + 2 * hi];
                av.y = A_s[i0 + row][kb + 2 * hi + 1];
                // B 4x16 frag: N striped over lanes, K over VGPRs/halves
                bv.x = M_s[kb + 2 * hi][a0 + row];
                bv.y = M_s[kb + 2 * hi + 1][a0 + row];
                c = __builtin_amdgcn_wmma_f32_16x16x4_f32(
                        false, av, false, bv, (short)0, c, false, false);
            }
            v8f yv;
            #pragma unroll
            for (int r = 0; r < 8; ++r) {
                const int gi = i0 + r + 8 * hi;      // C/D frag row mapping
                const int ga = a0 + row;
                const float Dv = d_v[gi] * dr_v[ga] /
                                 (fabsf(f_v[gi] - fr_v[ga]) + 1.0f);
                const float v = X_s[gi][ga] * Dv + c[r];
                yv[r] = v;
                ss += v * v;
            }
            return yv;
        };

        v8f y0 = doTile(t0);
        v8f y1 = {};
        if (two) y1 = doTile(t0 + NWAVES);   // wave-uniform branch: EXEC stays full

        // ---- phase 3: Frobenius norm (wave tree + 32-entry LDS combine) ----
        #pragma unroll
        for (int off = 16; off >= 1; off >>= 1)
            ss += __shfl_xor(ss, off, 32);
        if (lane == 0) partial[wave] = ss;
        __syncthreads();   // B3: partials ready; all X_s reads of this iter done

        float tot = 0.f;
        #pragma unroll
        for (int w = 0; w < NWAVES; ++w) tot += partial[w];
        const float inv = rsqrtf(tot);

        auto writeTile = [&](int t, const v8f& yv) {
            const int i0 = (t / 6) * 16;
            const int a0 = (t % 6) * 16;
            #pragma unroll
            for (int r = 0; r < 8; ++r)
                X_s[i0 + r + 8 * hi][a0 + row] = yv[r] * inv;
        };
        writeTile(t0, y0);
        if (two) writeTile(t0 + NWAVES, y1); // wave-uniform branch
    }

    __syncthreads();
    for (int e = tid; e < N * N; e += NT)
        out[e] = X_s[e / N][e % N];
}

extern "C" void kernel_launch(void* const* d_in, const int* in_sizes, int n_in,
                              void* d_out, int out_size, void* d_ws, size_t ws_size,
                              hipStream_t stream) {
    const float* adj       = (const float*)d_in[0];
    const float* adj_recon = (const float*)d_in[1];
    float* out = (float*)d_out;
    graphvae_mpm_kernel<<<dim3(1), dim3(NT), 0, stream>>>(adj, adj_recon, out);
}